// UserBookGNN_80590766342424
// MI455X (gfx1250) — compile-verified
//
#include <hip/hip_runtime.h>

#define NU 100000
#define NB 50000
#define NNODE (NU + NB)
#define H 128                 // HIDDEN
#define F 256                 // USER_FEAT == BOOK_FEAT
#define EPR 500000            // edges per relation
#define NE (4 * EPR)          // 2,000,000 directed edge slots (fwd + rev)

typedef __attribute__((ext_vector_type(2))) float v2f;
typedef __attribute__((ext_vector_type(8))) float v8f;

// ---------------------------------------------------------------------------
// GEMM: Y[M x H] = X[M x F] @ W[H x F]^T + bias   via V_WMMA_F32_16X16X4_F32
// One wave computes a 16-row x 128-col strip: the A fragment (v2f per lane)
// is reused across all 8 N-tiles, so X streams through HBM exactly once and
// W (128 KB) stays resident in L2/WGP$.
// A frag: lane L holds X[m0 + L%16][k + 2*(L/16) + j], j=0..1
// B frag: lane L holds W[n0 + L%16][k + 2*(L/16) + j]  (B = W^T tile)
// C/D:    VGPR v, lane L -> (M = v + 8*(L/16), N = L%16)
// ---------------------------------------------------------------------------
__global__ void __launch_bounds__(256)
gemm_bias_wmma(const float* __restrict__ X, const float* __restrict__ W,
               const float* __restrict__ bias, float* __restrict__ Y, int M) {
  const int wave = (int)((blockIdx.x * blockDim.x + threadIdx.x) >> 5);
  const int lane = threadIdx.x & 31;
  if (wave * 16 >= M) return;             // wave-uniform: EXEC stays all-ones
  const int mbase = wave * 16;
  const int lrow  = lane & 15;
  const int khalf = (lane >> 4) << 1;     // 0 or 2

  const float* arow = X + (size_t)(mbase + lrow) * F + khalf;

  v8f c[8] = {};                          // 8 accumulators = 64 VGPRs

  for (int k = 0; k < F; k += 4) {
    v2f a = *(const v2f*)(arow + k);
#pragma unroll
    for (int nt = 0; nt < 8; ++nt) {
      v2f b = *(const v2f*)(W + (size_t)(nt * 16 + lrow) * F + khalf + k);
      c[nt] = __builtin_amdgcn_wmma_f32_16x16x4_f32(
          /*neg_a=*/false, a, /*neg_b=*/false, b,
          /*c_mod=*/(short)0, c[nt], /*reuse_a=*/false, /*reuse_b=*/false);
    }
  }

  const int mo = (lane >> 4) << 3;        // 0 or 8
#pragma unroll
  for (int nt = 0; nt < 8; ++nt) {
    float bv = bias[nt * 16 + lrow];
#pragma unroll
    for (int v = 0; v < 8; ++v) {
      Y[(size_t)(mbase + mo + v) * H + nt * 16 + lrow] = c[nt][v] + bv;
    }
  }
}

// ---------------------------------------------------------------------------
// Edge decode matching the reference's full_edge ordering:
//   e in [0,      EPR)  : rates    fwd  row=u,      col=b
//   e in [EPR,  2*EPR)  : implicit fwd  row=u,      col=b
//   e in [2*EPR,3*EPR)  : rates    rev  row=b+NU,   col=u
//   e in [3*EPR,4*EPR)  : implicit rev  row=b+NU,   col=u
// Edge arrays are (2, EPR) row-major: [0][i]=user at i, [1][i]=book at EPR+i.
// ---------------------------------------------------------------------------
__device__ __forceinline__ void decode_edge(int e, const int* __restrict__ rates,
                                            const int* __restrict__ impl,
                                            int& row, int& col) {
  const int rev = (e >= 2 * EPR) ? 1 : 0;
  int idx = rev ? (e - 2 * EPR) : e;
  const int* ed = (idx < EPR) ? rates : impl;
  const int i = (idx < EPR) ? idx : (idx - EPR);
  const int u = ed[i];
  const int b = ed[i + EPR];
  row = rev ? (b + NU) : u;
  col = rev ? u : b;
}

__global__ void __launch_bounds__(256)
degree_kernel(const int* __restrict__ rates, const int* __restrict__ impl,
              float* __restrict__ deg) {
  const int e = blockIdx.x * blockDim.x + threadIdx.x;
  if (e >= NE) return;
  int row, col;
  decode_edge(e, rates, impl, row, col);
  unsafeAtomicAdd(deg + col, 1.0f);
}

__global__ void __launch_bounds__(256)
dis_kernel(const float* __restrict__ deg, float* __restrict__ dis) {
  const int i = blockIdx.x * blockDim.x + threadIdx.x;
  if (i >= NNODE) return;
  const float d = deg[i];
  dis[i] = (d > 0.0f) ? rsqrtf(d) : 0.0f;
}

__global__ void __launch_bounds__(256)
norm_kernel(const int* __restrict__ rates, const int* __restrict__ impl,
            const float* __restrict__ dis, float* __restrict__ nrm) {
  const int e = blockIdx.x * blockDim.x + threadIdx.x;
  if (e >= NE) return;
  int row, col;
  decode_edge(e, rates, impl, row, col);
  nrm[e] = dis[row] * dis[col];
}

// One wave per edge: float4 gather of x[row,:] (fully coalesced, 512 B/edge),
// scaled scatter into out[col,:] via hardware global_atomic_add_f32.
__global__ void __launch_bounds__(256)
scatter_kernel(const float* __restrict__ x, const float* __restrict__ nrm,
               const int* __restrict__ rates, const int* __restrict__ impl,
               float* __restrict__ out) {
  const int tid  = blockIdx.x * blockDim.x + threadIdx.x;
  const int e    = tid >> 5;
  const int lane = tid & 31;
  if (e >= NE) return;
  int row, col;
  decode_edge(e, rates, impl, row, col);
  const float n = nrm[e];
  const float4 v = ((const float4*)(x + (size_t)row * H))[lane];
  float* dst = out + (size_t)col * H + lane * 4;
  unsafeAtomicAdd(dst + 0, v.x * n);
  unsafeAtomicAdd(dst + 1, v.y * n);
  unsafeAtomicAdd(dst + 2, v.z * n);
  unsafeAtomicAdd(dst + 3, v.w * n);
}

// acc += buf (embedding-mean accumulator); x = buf (ping-pong for next layer)
__global__ void __launch_bounds__(256)
accum_copy_kernel(float* __restrict__ acc, float* __restrict__ x,
                  const float* __restrict__ buf) {
  const int i = blockIdx.x * blockDim.x + threadIdx.x;
  if (i >= NNODE * H) return;
  const float v = buf[i];
  acc[i] += v;
  x[i] = v;
}

__global__ void __launch_bounds__(256)
scale_kernel(float* __restrict__ acc) {
  const int i = blockIdx.x * blockDim.x + threadIdx.x;
  if (i >= NNODE * H) return;
  acc[i] *= (1.0f / 3.0f);   // mean over (NUM_LAYERS+1)=3 embedding snapshots
}

extern "C" void kernel_launch(void* const* d_in, const int* in_sizes, int n_in,
                              void* d_out, int out_size, void* d_ws, size_t ws_size,
                              hipStream_t stream) {
  (void)in_sizes; (void)n_in; (void)out_size; (void)ws_size;

  const float* user_x = (const float*)d_in[0];
  const float* book_x = (const float*)d_in[1];
  const float* user_W = (const float*)d_in[2];
  const float* user_b = (const float*)d_in[3];
  const float* book_W = (const float*)d_in[4];
  const float* book_b = (const float*)d_in[5];
  const int*   rates  = (const int*)d_in[6];
  const int*   impl   = (const int*)d_in[7];
  float* outp = (float*)d_out;             // [final_user | final_book] flat

  // Workspace carve-up (~163 MB total)
  float* x   = (float*)d_ws;               // NNODE*H  current embeddings
  float* buf = x   + (size_t)NNODE * H;    // NNODE*H  scatter target
  float* deg = buf + (size_t)NNODE * H;    // NNODE
  float* dis = deg + NNODE;                // NNODE
  float* nrm = dis + NNODE;                // NE

  const size_t embBytes = (size_t)NNODE * H * sizeof(float);
  hipMemsetAsync(deg, 0, NNODE * sizeof(float), stream);

  // Layer-0 projections (WMMA f32)
  {
    int waves = NU / 16;                                  // 6250, exact
    gemm_bias_wmma<<<(waves + 7) / 8, 256, 0, stream>>>(user_x, user_W, user_b, x, NU);
    waves = NB / 16;                                      // 3125, exact
    gemm_bias_wmma<<<(waves + 7) / 8, 256, 0, stream>>>(book_x, book_W, book_b,
                                                        x + (size_t)NU * H, NB);
  }

  // Accumulator starts as the layer-0 embeddings
  hipMemcpyAsync(outp, x, embBytes, hipMemcpyDeviceToDevice, stream);

  // Graph normalization
  degree_kernel<<<(NE + 255) / 256, 256, 0, stream>>>(rates, impl, deg);
  dis_kernel<<<(NNODE + 255) / 256, 256, 0, stream>>>(deg, dis);
  norm_kernel<<<(NE + 255) / 256, 256, 0, stream>>>(rates, impl, dis, nrm);

  // Two propagation layers
  const int nElem = NNODE * H;
  for (int layer = 0; layer < 2; ++layer) {
    hipMemsetAsync(buf, 0, embBytes, stream);
    const long long threads = (long long)NE * 32;
    scatter_kernel<<<(int)((threads + 255) / 256), 256, 0, stream>>>(x, nrm, rates,
                                                                     impl, buf);
    accum_copy_kernel<<<(nElem + 255) / 256, 256, 0, stream>>>(outp, x, buf);
  }

  scale_kernel<<<(nElem + 255) / 256, 256, 0, stream>>>(outp);
}